// MultiScaleFeatureAggregation_70952859730210
// MI455X (gfx1250) — compile-verified
//
#include <hip/hip_runtime.h>
#include <hip/hip_bf16.h>

typedef float v2f __attribute__((ext_vector_type(2)));
typedef float v8f __attribute__((ext_vector_type(8)));

#define NPTS    2048
#define NBATCH  4
#define NCOL    64     // output columns per workgroup
#define BSTRIDE 80     // LDS row stride for B-operand buffers: banks (16k+col)%64
                       // give lanes 0-15 / 16-31 disjoint bank sets -> conflict-free

// Full-precision f32 WMMA: D = A(16x4) * B(4x16) + C, exact FMA chain like the
// f32 reference. 8-arg form: (neg_a, A, neg_b, B, c_mod, C, reuse_a, reuse_b).
__device__ __forceinline__ v8f wmma_f32_k4(v2f a, v2f b, v8f c) {
  return __builtin_amdgcn_wmma_f32_16x16x4_f32(false, a, false, b, (short)0, c,
                                               false, false);
}

// relu(W[CROWS x K] @ B_lds[K x NCOL] + bias) -> LDS or global.
// A fragments: lane l holds rows rt*16+(l&15), K pair (l>=16 ? {2,3} : {0,1}).
// W is row-major in global memory (L2-resident, reused by all 128 blocks).
// `wave` is an SGPR (readfirstlane) so the rt loop is a uniform scalar loop:
// EXEC stays all-ones across every WMMA, no exec-mask structurization.
template <int K, int CROWS, bool TO_LDS>
__device__ __forceinline__ void gemm_stage(const float* __restrict__ W,
                                           const float* __restrict__ bias,
                                           const float* sB,     // LDS [K][BSTRIDE]
                                           float* sOut,         // LDS [CROWS][BSTRIDE]
                                           float* __restrict__ gOut, // out+(b*1024)*2048+colBase
                                           int lane, int wave) {
  const int m15   = lane & 15;
  const int khalf = (lane >> 4) << 1;  // 0 or 2
  const int rsel  = (lane >> 4) << 3;  // D rows: vgpr r -> M = r (+8 for upper lanes)
  constexpr int NK = K / 4;

#pragma unroll 1
  for (int rt = wave; rt < CROWS / 16; rt += 8) {
    v2f afr[NK];
    const float* wp = W + (size_t)(rt * 16 + m15) * K + khalf;
#pragma unroll
    for (int kk = 0; kk < NK; ++kk)
      afr[kk] = *(const v2f*)(wp + kk * 4);  // 8B-aligned strided load, L2 hit

#pragma unroll
    for (int ct = 0; ct < 4; ++ct) {
      v8f acc = {};
      const float* bp = sB + ct * 16 + m15 + khalf * BSTRIDE;
#pragma unroll
      for (int kk = 0; kk < NK; ++kk) {
        v2f b;
        b.x = bp[(kk * 4 + 0) * BSTRIDE];
        b.y = bp[(kk * 4 + 1) * BSTRIDE];
        acc = wmma_f32_k4(afr[kk], b, acc);
      }
#pragma unroll
      for (int r = 0; r < 8; ++r) {
        const int row = rt * 16 + r + rsel;
        float v = acc[r] + bias[row];
        v = v > 0.0f ? v : 0.0f;
        if (TO_LDS)
          sOut[row * BSTRIDE + ct * 16 + m15] = v;
        else
          gOut[row * NPTS + ct * 16 + m15] = v;  // lanes 0-15 / 16-31: two 64B runs
      }
    }
  }
}

__global__ __launch_bounds__(256) void msfa_fusion_kernel(
    const float* __restrict__ x,
    const float* __restrict__ W1, const float* __restrict__ W2,
    const float* __restrict__ W3, const float* __restrict__ b1,
    const float* __restrict__ b2, const float* __restrict__ b3,
    float* __restrict__ out) {
  __shared__ float sXp[4 * BSTRIDE];     // X tile, K padded 3->4
  __shared__ float sW1[64 * 4];          // W1 padded 64x4
  __shared__ float sC1[64 * BSTRIDE];    // L1 activations
  __shared__ float sC2[128 * BSTRIDE];   // L2 activations

  const int tid     = threadIdx.x;
  const int lane    = tid & 31;
  // Wave index is lane-invariant in wave32; pin it to an SGPR so all per-wave
  // loops compile to uniform scalar control flow (no exec masking).
  const int wave    = __builtin_amdgcn_readfirstlane(tid >> 5);
  const int batch   = blockIdx.x >> 5;        // 32 column tiles per batch
  const int colBase = (blockIdx.x & 31) * NCOL;

  // Warm L2 with W3 (512 KB, reused by every block): 8 x 256B-stride
  // prefetches per thread tile the whole tensor. Constant trip count.
#pragma unroll
  for (int i = 0; i < 8; ++i)
    __builtin_prefetch(W3 + (size_t)(tid + i * 256) * 64, 0, 3);

  // Stage X tile (zero-pad K row 3): 256 threads cover 4x64 exactly.
  {
    const int c = tid >> 6, col = tid & 63;
    sXp[c * BSTRIDE + col] =
        (c < 3) ? x[(size_t)(batch * 3 + c) * NPTS + colBase + col] : 0.0f;
  }
  // Stage W1, zero-padded to 64x4: 256 threads cover 64x4 exactly.
  {
    const int m = tid >> 2, k = tid & 3;
    sW1[tid] = (k < 3) ? W1[m * 3 + k] : 0.0f;
  }
  __syncthreads();

  // ---- Layer 1: relu(W1[64x3] @ X[3x64] + b1) -> sC1, one WMMA per 16x16 tile
  {
    const int m15   = lane & 15;
    const int khalf = (lane >> 4) << 1;
    const int rsel  = (lane >> 4) << 3;
#pragma unroll
    for (int i = 0; i < 2; ++i) {        // tiles t = wave, wave+8 (constant trips)
      const int t = wave + i * 8;
      const int rt = t >> 2, ct = t & 3;
      v2f a, b;
      a.x = sW1[(rt * 16 + m15) * 4 + khalf];
      a.y = sW1[(rt * 16 + m15) * 4 + khalf + 1];
      b.x = sXp[(khalf + 0) * BSTRIDE + ct * 16 + m15];
      b.y = sXp[(khalf + 1) * BSTRIDE + ct * 16 + m15];
      v8f acc = {};
      acc = wmma_f32_k4(a, b, acc);
#pragma unroll
      for (int r = 0; r < 8; ++r) {
        const int row = rt * 16 + r + rsel;
        float v = acc[r] + b1[row];
        sC1[row * BSTRIDE + ct * 16 + m15] = v > 0.0f ? v : 0.0f;
      }
    }
  }
  __syncthreads();

  // ---- Layer 2: relu(W2[128x64] @ C1 + b2) -> sC2
  gemm_stage<64, 128, true>(W2, b2, sC1, sC2, nullptr, lane, wave);
  __syncthreads();

  // ---- Layer 3: relu(W3[1024x128] @ C2 + b3) -> global (single HBM pass)
  float* gOut = out + (size_t)batch * 1024 * NPTS + colBase;
  gemm_stage<128, 1024, false>(W3, b3, sC2, nullptr, gOut, lane, wave);
}

extern "C" void kernel_launch(void* const* d_in, const int* in_sizes, int n_in,
                              void* d_out, int out_size, void* d_ws,
                              size_t ws_size, hipStream_t stream) {
  // p0/p1/p2 in the reference are dead (del'd) -> only the fusion MLP matters.
  // Locate fusion weights: W3 is the unique 1024*128 = 131072-element input;
  // leaves are ordered ... fusion_W1, fusion_W2, fusion_W3, fusion_b1..b3.
  int i3 = -1;
  for (int i = 0; i < n_in; ++i)
    if (in_sizes[i] == 1024 * 128) { i3 = i; break; }
  if (i3 < 2 || i3 + 3 >= n_in) i3 = 21;  // setup_inputs() dict-order fallback

  const float* x  = (const float*)d_in[0];
  const float* W1 = (const float*)d_in[i3 - 2];
  const float* W2 = (const float*)d_in[i3 - 1];
  const float* W3 = (const float*)d_in[i3];
  const float* b1 = (const float*)d_in[i3 + 1];
  const float* b2 = (const float*)d_in[i3 + 2];
  const float* b3 = (const float*)d_in[i3 + 3];

  msfa_fusion_kernel<<<dim3(NBATCH * NPTS / NCOL), dim3(256), 0, stream>>>(
      x, W1, W2, W3, b1, b2, b3, (float*)d_out);
}